// GIN_55585466744867
// MI455X (gfx1250) — compile-verified
//
#include <hip/hip_runtime.h>

typedef __attribute__((ext_vector_type(2))) float v2f;
typedef __attribute__((ext_vector_type(8))) float v8f;

#define D_IN 64
#define HID  128
#define N_GRAPHS 256

// ---------------------------------------------------------------------------
// Elementwise copy (float4 vectorized), n must be divisible by 4.
__global__ void gin_copy4(float* __restrict__ dst, const float* __restrict__ src, long long n4) {
    long long gid = (long long)blockIdx.x * blockDim.x + threadIdx.x;
    if (gid >= n4) return;
    ((float4*)dst)[gid] = ((const float4*)src)[gid];
}

// Zero-fill
__global__ void gin_zero(float* __restrict__ p, long long n) {
    long long gid = (long long)blockIdx.x * blockDim.x + threadIdx.x;
    if (gid < n) p[gid] = 0.0f;
}

// ---------------------------------------------------------------------------
// Edge scatter-add: acc[dst[e]][:] += feat[src[e]][:]  (F floats per row,
// 4 floats per thread). Rows are 16B aligned so float4 gather loads are fine.
__global__ void gin_scatter_add(float* __restrict__ acc, const float* __restrict__ feat,
                                const int* __restrict__ src, const int* __restrict__ dst,
                                long long E, int chunks, int F) {
    long long gid = (long long)blockIdx.x * blockDim.x + threadIdx.x;
    long long total = E * (long long)chunks;
    if (gid >= total) return;
    int e = (int)(gid / chunks);
    int c = (int)(gid % chunks);
    int s = src[e];
    int d = dst[e];
    float4 v = *(const float4*)(feat + (size_t)s * F + c * 4);
    float* p = acc + (size_t)d * F + c * 4;
    atomicAdd(p + 0, v.x);
    atomicAdd(p + 1, v.y);
    atomicAdd(p + 2, v.z);
    atomicAdd(p + 3, v.w);
}

// ---------------------------------------------------------------------------
// out[M,128] = relu(in[M,K] @ W[K,128] + bias[128])
// One wave computes a 16x16 tile with V_WMMA_F32_16X16X4_F32 (fp32-exact).
// 8 waves / block -> block covers 16 rows x 128 cols. K in {64,128}.
__global__ __launch_bounds__(256) void gin_gemm_bias_relu(
    float* __restrict__ out, const float* __restrict__ in,
    const float* __restrict__ W, const float* __restrict__ bias,
    int K, int M) {
    const int lane = threadIdx.x & 31;
    const int wave = threadIdx.x >> 5;
    const int row0 = blockIdx.x * 16;
    const int col0 = wave * 16;
    const int half = lane >> 4;   // 0: lanes 0-15, 1: lanes 16-31
    const int l16  = lane & 15;
    const int koff = half * 2;    // ISA A/B layout: VGPR0 holds K=0|K=2, VGPR1 K=1|K=3
    int arow = row0 + l16;
    if (arow >= M) arow = M - 1;  // clamp loads; EXEC stays all-ones for WMMA
    const int bcol = col0 + l16;

    v8f acc = {};
    for (int k = 0; k < K; k += 4) {
        v2f a = *(const v2f*)(in + (size_t)arow * K + k + koff);
        v2f b;
        b.x = W[(size_t)(k + koff)     * HID + bcol];
        b.y = W[(size_t)(k + koff + 1) * HID + bcol];
        acc = __builtin_amdgcn_wmma_f32_16x16x4_f32(
            /*neg_a=*/false, a, /*neg_b=*/false, b,
            /*c_mod=*/(short)0, acc, /*reuse_a=*/false, /*reuse_b=*/false);
    }

    const float bv = bias[bcol];
    // C/D layout: VGPR v -> row (row0 + v + 8*half), col bcol
    float* obase = out + (size_t)(row0 + half * 8) * HID + bcol;
#pragma unroll
    for (int v = 0; v < 8; ++v) {
        int r = row0 + half * 8 + v;
        if (r < M) {
            float x = acc[v] + bv;
            obase[(size_t)v * HID] = x > 0.0f ? x : 0.0f;
        }
    }
}

// ---------------------------------------------------------------------------
// Mean-pool part 1: pooled[batch[n]][:] += h[n][:]
__global__ void gin_pool_add(float* __restrict__ pooled, const float* __restrict__ h,
                             const int* __restrict__ batch, long long nodes) {
    long long gid = (long long)blockIdx.x * blockDim.x + threadIdx.x;
    long long total = nodes * (HID / 4);
    if (gid >= total) return;
    int n = (int)(gid / (HID / 4));
    int c = (int)(gid % (HID / 4));
    int g = batch[n];
    float4 v = *(const float4*)(h + (size_t)n * HID + c * 4);
    float* p = pooled + (size_t)g * HID + c * 4;
    atomicAdd(p + 0, v.x);
    atomicAdd(p + 1, v.y);
    atomicAdd(p + 2, v.z);
    atomicAdd(p + 3, v.w);
}

// Mean-pool part 2: counts
__global__ void gin_count(float* __restrict__ counts, const int* __restrict__ batch,
                          long long nodes) {
    long long gid = (long long)blockIdx.x * blockDim.x + threadIdx.x;
    if (gid >= nodes) return;
    atomicAdd(&counts[batch[gid]], 1.0f);
}

// Final FC: out[g] = dot(pooled[g]/max(count,1), Wfc) + bfc
__global__ __launch_bounds__(128) void gin_final_fc(
    float* __restrict__ out, const float* __restrict__ pooled,
    const float* __restrict__ counts, const float* __restrict__ Wfc,
    const float* __restrict__ bfc) {
    const int g = blockIdx.x;
    const int t = threadIdx.x;
    float cnt = counts[g];
    cnt = cnt > 1.0f ? cnt : 1.0f;
    float val = (pooled[(size_t)g * HID + t] / cnt) * Wfc[t];
#pragma unroll
    for (int off = 16; off > 0; off >>= 1)
        val += __shfl_down(val, off, 32);
    __shared__ float s[4];
    if ((t & 31) == 0) s[t >> 5] = val;
    __syncthreads();
    if (t == 0) out[g] = s[0] + s[1] + s[2] + s[3] + bfc[0];
}

// ---------------------------------------------------------------------------
extern "C" void kernel_launch(void* const* d_in, const int* in_sizes, int n_in,
                              void* d_out, int out_size, void* d_ws, size_t ws_size,
                              hipStream_t stream) {
    const float* x    = (const float*)d_in[0];
    const int*   ei   = (const int*)d_in[1];   // [2, E]: row0 = src, row1 = dst
    const int*   bat  = (const int*)d_in[2];
    const float* W1a  = (const float*)d_in[3];
    const float* b1a  = (const float*)d_in[4];
    const float* W1b  = (const float*)d_in[5];
    const float* b1b  = (const float*)d_in[6];
    const float* W2a  = (const float*)d_in[7];
    const float* b2a  = (const float*)d_in[8];
    const float* W2b  = (const float*)d_in[9];
    const float* b2b  = (const float*)d_in[10];
    const float* Wfc  = (const float*)d_in[11];
    const float* bfc  = (const float*)d_in[12];
    float* out = (float*)d_out;

    const long long nodes = (long long)in_sizes[0] / D_IN;
    const long long E     = (long long)in_sizes[1] / 2;
    const int* src = ei;
    const int* dst = ei + E;

    float* Z = (float*)d_ws;                       // aggregation / GEMM input
    float* T = Z + (size_t)nodes * HID;            // MLP hidden
    float* H = T + (size_t)nodes * HID;            // layer output
    float* pooled = H + (size_t)nodes * HID;       // [256,128]
    float* counts = pooled + (size_t)N_GRAPHS * HID;

    const int B = 256;
    const int mblocks = (int)((nodes + 15) / 16);
    auto blks = [](long long n, int b) { return (unsigned)((n + b - 1) / b); };

    // ---- Layer 1: Z = x; Z[dst] += x[src]; T = relu(Z@W1a+b1a); H = relu(T@W1b+b1b)
    gin_copy4<<<blks(nodes * D_IN / 4, B), B, 0, stream>>>(Z, x, nodes * D_IN / 4);
    gin_scatter_add<<<blks(E * (D_IN / 4), B), B, 0, stream>>>(Z, x, src, dst, E, D_IN / 4, D_IN);
    gin_gemm_bias_relu<<<mblocks, B, 0, stream>>>(T, Z, W1a, b1a, D_IN, (int)nodes);
    gin_gemm_bias_relu<<<mblocks, B, 0, stream>>>(H, T, W1b, b1b, HID, (int)nodes);

    // ---- Layer 2: Z = H; Z[dst] += H[src]; T = relu(Z@W2a+b2a); H = relu(T@W2b+b2b)
    gin_copy4<<<blks(nodes * HID / 4, B), B, 0, stream>>>(Z, H, nodes * HID / 4);
    gin_scatter_add<<<blks(E * (HID / 4), B), B, 0, stream>>>(Z, H, src, dst, E, HID / 4, HID);
    gin_gemm_bias_relu<<<mblocks, B, 0, stream>>>(T, Z, W2a, b2a, HID, (int)nodes);
    gin_gemm_bias_relu<<<mblocks, B, 0, stream>>>(H, T, W2b, b2b, HID, (int)nodes);

    // ---- Global mean pool + final FC
    const long long pn = (long long)N_GRAPHS * HID + N_GRAPHS;
    gin_zero<<<blks(pn, B), B, 0, stream>>>(pooled, pn);
    gin_pool_add<<<blks(nodes * (HID / 4), B), B, 0, stream>>>(pooled, H, bat, nodes);
    gin_count<<<blks(nodes, B), B, 0, stream>>>(counts, bat, nodes);
    gin_final_fc<<<N_GRAPHS, HID, 0, stream>>>(out, pooled, counts, Wfc, bfc);

    (void)n_in; (void)out_size; (void)ws_size;
}